// GraphTransformerBlock_75557064671477
// MI455X (gfx1250) — compile-verified
//
#include <hip/hip_runtime.h>

// ---------------- problem constants (match reference) ----------------
#define N_NODES 50000
#define DIM_IN  128
#define DIM_OUT 128
#define HEADS   3
#define EDIM    16
#define NEDGE   800000
#define NEDGE2  (NEDGE + N_NODES)          // edges + self loops
#define HC      (HEADS * DIM_OUT)          // 384
#define NEG_SLOPE 0.2f
#define LN_EPS    1e-5f

typedef __attribute__((ext_vector_type(2))) float v2f;
typedef __attribute__((ext_vector_type(8))) float v8f;

// ---------------- workspace layout (float offsets) ----------------
// accumulators (zeroed every call) first, as one contiguous region:
#define OFF_AGG   0L                                   // N*384
#define OFF_DEG   (OFF_AGG   + (long)N_NODES*HC)       // N
#define OFF_SATTR (OFF_DEG   + (long)N_NODES)          // N*16
#define OFF_MMAX  (OFF_SATTR + (long)N_NODES*EDIM)     // N*3 (as uint)
#define OFF_DENOM (OFF_MMAX  + (long)N_NODES*HEADS)    // N*3
#define ZERO_COUNT (OFF_DENOM + (long)N_NODES*HEADS)   // end of zero region
#define OFF_XP    ZERO_COUNT                           // N*384
#define OFF_Y     (OFF_XP    + (long)N_NODES*HC)       // N*128
#define OFF_ASRC  (OFF_Y     + (long)N_NODES*DIM_OUT)  // N*3
#define OFF_ADST  (OFF_ASRC  + (long)N_NODES*HEADS)    // N*3
#define OFF_LATTR (OFF_ADST  + (long)N_NODES*HEADS)    // N*16
#define OFF_ALPHA (OFF_LATTR + (long)N_NODES*EDIM)     // E2*3 (alpha, then exp in place)
#define OFF_MEDGE (OFF_ALPHA + (long)NEDGE2*HEADS)     // 64
#define OFF_BIAS2 (OFF_MEDGE + 64L)                    // 128

// order-preserving float <-> uint mapping for atomic max
__device__ __forceinline__ unsigned fmap_ord(float f) {
  unsigned u = __float_as_uint(f);
  return (f >= 0.0f) ? (u | 0x80000000u) : ~u;
}
__device__ __forceinline__ float funmap_ord(unsigned m) {
  return (m & 0x80000000u) ? __uint_as_float(m & 0x7FFFFFFFu) : __uint_as_float(~m);
}

// ---------------- K0: zero the accumulator region ----------------
__global__ void k_zero(float* __restrict__ p, long n) {
  long i = (long)blockIdx.x * blockDim.x + threadIdx.x;
  if (i < n) p[i] = 0.0f;
}

// ---------------- K1: deg + per-dst sum of edge_attr ----------------
__global__ void k_edge_accum(const long long* __restrict__ col,
                             const float* __restrict__ ea,
                             float* __restrict__ deg,
                             float* __restrict__ sattr) {
  int t = blockIdx.x * blockDim.x + threadIdx.x;
  if (t >= NEDGE * EDIM) return;
  int e = t >> 4, d = t & 15;
  long long c = col[e];
  atomicAdd(&sattr[c * EDIM + d], ea[t]);
  if (d == 0) atomicAdd(&deg[c], 1.0f);
}

// ---------------- K2: loop_attr = sum/deg ----------------
__global__ void k_loop_attr(const float* __restrict__ sattr,
                            const float* __restrict__ deg,
                            float* __restrict__ lattr) {
  int t = blockIdx.x * blockDim.x + threadIdx.x;
  if (t >= N_NODES * EDIM) return;
  lattr[t] = sattr[t] / fmaxf(deg[t >> 4], 1.0f);
}

// ---------------- K3: tiny precomputes: M_edge[16][3], bias2[128] ----------------
__global__ void k_prep(const float* __restrict__ W_edge, const float* __restrict__ att_edge,
                       const float* __restrict__ gat_bias, const float* __restrict__ lin_W,
                       const float* __restrict__ lin_b,
                       float* __restrict__ M_edge, float* __restrict__ bias2) {
  int t = threadIdx.x;  // 128 threads, one block
  float s = lin_b[t];
  for (int k = 0; k < HC; ++k) s += gat_bias[k] * lin_W[k * DIM_OUT + t];
  bias2[t] = s;
  if (t < EDIM * HEADS) {
    int d = t / HEADS, h = t % HEADS;
    float m = 0.0f;
    for (int c = 0; c < DIM_OUT; ++c)
      m += W_edge[d * HC + h * DIM_OUT + c] * att_edge[h * DIM_OUT + c];
    M_edge[d * HEADS + h] = m;
  }
}

// ---------------- WMMA fp32 GEMM: D[M,N] = A[M,K] * B[K,N] ----------------
// row-major; M mult of 16, N mult of 64, K mult of 4.
// One wave -> one 16x64 tile (4 independent accumulators; A fragment reused 4x).
// V_WMMA_F32_16X16X4_F32 fragment layout (ISA 7.12.2):
//   A 16x4 : lanes 0-15 row m=lane, v0=K0,v1=K1; lanes 16-31 row m=lane-16, v0=K2,v1=K3
//   B 4x16 : lanes 0-15 col n=lane, v0=row K0,v1=row K1; lanes 16-31 v0=K2,v1=K3
//   C/D    : vgpr g: lanes 0-15 -> D[g, lane]; lanes 16-31 -> D[g+8, lane-16]
__global__ void k_wmma_gemm_f32(const float* __restrict__ A, const float* __restrict__ B,
                                float* __restrict__ D, int M, int N, int K) {
  int wave = (blockIdx.x * blockDim.x + threadIdx.x) >> 5;
  int lane = threadIdx.x & 31;
  int ncg = N >> 6;                      // column groups of 64
  int tm = wave / ncg, tg = wave - tm * ncg;
  if (tm >= (M >> 4)) return;            // wave-uniform exit: EXEC all-1 for WMMA
  int m0 = tm << 4, n0 = tg << 6;
  int half = lane >> 4;                  // 0: lanes 0-15, 1: lanes 16-31
  int l16 = lane & 15;

  v8f acc0 = {}, acc1 = {}, acc2 = {}, acc3 = {};
  const float* Ap = A + (long)(m0 + l16) * K + (half << 1);   // 8B aligned -> b64 loads
  const float* Bp = B + (long)(half << 1) * N + n0 + l16;

#pragma unroll 2
  for (int k = 0; k < K; k += 4) {
    v2f a = *(const v2f*)Ap;
    v2f b0, b1, b2, b3;
    b0.x = Bp[0];   b0.y = Bp[N];
    b1.x = Bp[16];  b1.y = Bp[N + 16];
    b2.x = Bp[32];  b2.y = Bp[N + 32];
    b3.x = Bp[48];  b3.y = Bp[N + 48];
    acc0 = __builtin_amdgcn_wmma_f32_16x16x4_f32(false, a, false, b0, (short)0, acc0, false, false);
    acc1 = __builtin_amdgcn_wmma_f32_16x16x4_f32(false, a, false, b1, (short)0, acc1, false, false);
    acc2 = __builtin_amdgcn_wmma_f32_16x16x4_f32(false, a, false, b2, (short)0, acc2, false, false);
    acc3 = __builtin_amdgcn_wmma_f32_16x16x4_f32(false, a, false, b3, (short)0, acc3, false, false);
    Ap += 4;
    Bp += (long)4 * N;
  }

  float* Dp = D + (long)(m0 + half * 8) * N + n0 + l16;
#pragma unroll
  for (int g = 0; g < 8; ++g) {
    float* r = Dp + (long)g * N;
    r[0]  = acc0[g];
    r[16] = acc1[g];
    r[32] = acc2[g];
    r[48] = acc3[g];
  }
}

// ---------------- K4: a_src/a_dst = <xp[n,h,:], att_*[h,:]> (one wave per (n,h)) ----------------
__global__ void k_attn_node(const float* __restrict__ xp,
                            const float* __restrict__ att_src, const float* __restrict__ att_dst,
                            float* __restrict__ a_src, float* __restrict__ a_dst) {
  int gw = (blockIdx.x * blockDim.x + threadIdx.x) >> 5;
  int lane = threadIdx.x & 31;
  if (gw >= N_NODES * HEADS) return;
  int n = gw / HEADS, h = gw - n * HEADS;
  const float* xph = xp + (long)n * HC + h * DIM_OUT;
  float s = 0.0f, d = 0.0f;
#pragma unroll
  for (int i = 0; i < DIM_OUT; i += 32) {
    float v = xph[i + lane];
    s += v * att_src[h * DIM_OUT + i + lane];
    d += v * att_dst[h * DIM_OUT + i + lane];
  }
#pragma unroll
  for (int o = 16; o; o >>= 1) {
    s += __shfl_xor(s, o, 32);
    d += __shfl_xor(d, o, 32);
  }
  if (lane == 0) { a_src[gw] = s; a_dst[gw] = d; }
}

// ---------------- K5: alpha = leakyrelu(a_src[r]+a_dst[c]+ea.M_edge); atomic segment max ------
__global__ void k_alpha_max(const long long* __restrict__ row, const long long* __restrict__ col,
                            const float* __restrict__ ea, const float* __restrict__ lattr,
                            const float* __restrict__ a_src, const float* __restrict__ a_dst,
                            const float* __restrict__ M_edge,
                            float* __restrict__ alpha, unsigned* __restrict__ mmax) {
  int e = blockIdx.x * blockDim.x + threadIdx.x;
  if (e >= NEDGE2) return;
  long long r, c;
  const float* attr;
  if (e < NEDGE) { r = row[e]; c = col[e]; attr = ea + (long)e * EDIM; }
  else           { r = c = e - NEDGE;      attr = lattr + (long)(e - NEDGE) * EDIM; }
  float ae0 = 0.0f, ae1 = 0.0f, ae2 = 0.0f;
#pragma unroll
  for (int d = 0; d < EDIM; ++d) {
    float v = attr[d];
    ae0 += v * M_edge[d * HEADS + 0];
    ae1 += v * M_edge[d * HEADS + 1];
    ae2 += v * M_edge[d * HEADS + 2];
  }
  float aeh[HEADS] = {ae0, ae1, ae2};
#pragma unroll
  for (int h = 0; h < HEADS; ++h) {
    float al = a_src[r * HEADS + h] + a_dst[c * HEADS + h] + aeh[h];
    al = (al >= 0.0f) ? al : NEG_SLOPE * al;
    alpha[(long)e * HEADS + h] = al;
    atomicMax(&mmax[c * HEADS + h], fmap_ord(al));
  }
}

// ---------------- K6: exp(alpha - max) in place; atomic segment sum -> denom ----------------
__global__ void k_exp_denom(const long long* __restrict__ col,
                            const unsigned* __restrict__ mmax,
                            float* __restrict__ alpha, float* __restrict__ denom) {
  int e = blockIdx.x * blockDim.x + threadIdx.x;
  if (e >= NEDGE2) return;
  long long c = (e < NEDGE) ? col[e] : (long long)(e - NEDGE);
#pragma unroll
  for (int h = 0; h < HEADS; ++h) {
    float m = funmap_ord(mmax[c * HEADS + h]);
    float ex = __expf(alpha[(long)e * HEADS + h] - m);
    alpha[(long)e * HEADS + h] = ex;
    atomicAdd(&denom[c * HEADS + h], ex);
  }
}

// ---------------- K7: scatter messages: agg[c,h,:] += (ex/denom) * xp[r,h,:] ----------------
// 96 threads per edge: (h in 0..2) x (32 lanes), each lane moves a float4.
__global__ void k_scatter(const long long* __restrict__ row, const long long* __restrict__ col,
                          const float* __restrict__ alpha, const float* __restrict__ denom,
                          const float* __restrict__ xp, float* __restrict__ agg) {
  int t = blockIdx.x * blockDim.x + threadIdx.x;
  int e = t / 96;
  if (e >= NEDGE2) return;
  int q = t - e * 96;
  int h = q >> 5, lane = q & 31;
  long long r, c;
  if (e < NEDGE) { r = row[e]; c = col[e]; }
  else           { r = c = e - NEDGE; }
  float w = alpha[(long)e * HEADS + h] / denom[c * HEADS + h];
  const float4 xv = *(const float4*)(xp + r * HC + h * DIM_OUT + (lane << 2));
  float* dst = agg + c * HC + h * DIM_OUT + (lane << 2);
  atomicAdd(dst + 0, w * xv.x);
  atomicAdd(dst + 1, w * xv.y);
  atomicAdd(dst + 2, w * xv.z);
  atomicAdd(dst + 3, w * xv.w);
}

// ---------------- K8: h = x + y + bias2; LayerNorm over 128; one wave per node -------------
__global__ void k_residual_ln(const float* __restrict__ x, const float* __restrict__ y,
                              const float* __restrict__ bias2,
                              const float* __restrict__ ln_g, const float* __restrict__ ln_b,
                              float* __restrict__ out) {
  int n = (blockIdx.x * blockDim.x + threadIdx.x) >> 5;
  int lane = threadIdx.x & 31;
  if (n >= N_NODES) return;
  float hv[4];
  float s = 0.0f;
#pragma unroll
  for (int i = 0; i < 4; ++i) {
    int c = lane + i * 32;
    hv[i] = x[(long)n * DIM_OUT + c] + y[(long)n * DIM_OUT + c] + bias2[c];
    s += hv[i];
  }
#pragma unroll
  for (int o = 16; o; o >>= 1) s += __shfl_xor(s, o, 32);
  float mu = s * (1.0f / DIM_OUT);
  float vs = 0.0f;
#pragma unroll
  for (int i = 0; i < 4; ++i) { float d = hv[i] - mu; vs += d * d; }
#pragma unroll
  for (int o = 16; o; o >>= 1) vs += __shfl_xor(vs, o, 32);
  float inv = rsqrtf(vs * (1.0f / DIM_OUT) + LN_EPS);
#pragma unroll
  for (int i = 0; i < 4; ++i) {
    int c = lane + i * 32;
    out[(long)n * DIM_OUT + c] = (hv[i] - mu) * inv * ln_g[c] + ln_b[c];
  }
}

// =====================================================================
extern "C" void kernel_launch(void* const* d_in, const int* in_sizes, int n_in,
                              void* d_out, int out_size, void* d_ws, size_t ws_size,
                              hipStream_t stream) {
  const float*     x        = (const float*)d_in[0];
  const long long* ei       = (const long long*)d_in[1];   // jnp.int64
  const float*     ea       = (const float*)d_in[2];
  const float*     W        = (const float*)d_in[3];
  const float*     W_edge   = (const float*)d_in[4];
  const float*     att_src  = (const float*)d_in[5];
  const float*     att_dst  = (const float*)d_in[6];
  const float*     att_edge = (const float*)d_in[7];
  const float*     gat_bias = (const float*)d_in[8];
  const float*     lin_W    = (const float*)d_in[9];
  const float*     lin_b    = (const float*)d_in[10];
  const float*     ln_g     = (const float*)d_in[11];
  const float*     ln_b     = (const float*)d_in[12];
  float* out = (float*)d_out;

  const long long* row = ei;
  const long long* col = ei + NEDGE;

  float* ws = (float*)d_ws;
  float*    agg   = ws + OFF_AGG;
  float*    deg   = ws + OFF_DEG;
  float*    sattr = ws + OFF_SATTR;
  unsigned* mmax  = (unsigned*)(ws + OFF_MMAX);
  float*    denom = ws + OFF_DENOM;
  float*    xp    = ws + OFF_XP;
  float*    yv    = ws + OFF_Y;
  float*    a_src = ws + OFF_ASRC;
  float*    a_dst = ws + OFF_ADST;
  float*    lattr = ws + OFF_LATTR;
  float*    alpha = ws + OFF_ALPHA;
  float*    Medge = ws + OFF_MEDGE;
  float*    bias2 = ws + OFF_BIAS2;

  const int B = 256;

  // K0: zero all accumulators
  {
    long n = ZERO_COUNT;
    k_zero<<<(int)((n + B - 1) / B), B, 0, stream>>>(ws, n);
  }
  // K1: degree + summed edge_attr per destination
  k_edge_accum<<<(NEDGE * EDIM + B - 1) / B, B, 0, stream>>>(col, ea, deg, sattr);
  // K2: self-loop attrs (per-dst mean)
  k_loop_attr<<<(N_NODES * EDIM + B - 1) / B, B, 0, stream>>>(sattr, deg, lattr);
  // K3: M_edge (folds W_edge*att_edge) and bias2 (folds gat_bias@lin_W + lin_b)
  k_prep<<<1, 128, 0, stream>>>(W_edge, att_edge, gat_bias, lin_W, lin_b, Medge, bias2);
  // GEMM1: xp = x @ W   (50000 x 128 x 384), 16x64 tile per wave
  {
    int waves = (N_NODES / 16) * (HC / 64);
    k_wmma_gemm_f32<<<(waves * 32 + B - 1) / B, B, 0, stream>>>(x, W, xp, N_NODES, HC, DIM_IN);
  }
  // K4: per-node attention terms
  k_attn_node<<<(N_NODES * HEADS * 32 + B - 1) / B, B, 0, stream>>>(xp, att_src, att_dst, a_src, a_dst);
  // K5: edge logits + segment max
  k_alpha_max<<<(NEDGE2 + B - 1) / B, B, 0, stream>>>(row, col, ea, lattr, a_src, a_dst, Medge, alpha, mmax);
  // K6: exp + segment sum
  k_exp_denom<<<(NEDGE2 + B - 1) / B, B, 0, stream>>>(col, mmax, alpha, denom);
  // K7: weighted message scatter-add (the HBM-bound phase)
  {
    long t = (long)NEDGE2 * 96;
    k_scatter<<<(int)((t + B - 1) / B), B, 0, stream>>>(row, col, alpha, denom, xp, agg);
  }
  // GEMM2: y = agg @ lin_W  (50000 x 384 x 128), 16x64 tile per wave
  {
    int waves = (N_NODES / 16) * (DIM_OUT / 64);
    k_wmma_gemm_f32<<<(waves * 32 + B - 1) / B, B, 0, stream>>>(agg, lin_W, yv, N_NODES, DIM_OUT, HC);
  }
  // K8: residual + LayerNorm
  k_residual_ln<<<(N_NODES * 32 + B - 1) / B, B, 0, stream>>>(x, yv, bias2, ln_g, ln_b, out);
}